// SSD_GMM_86517821215618
// MI455X (gfx1250) — compile-verified
//
#include <hip/hip_runtime.h>
#include <cstdint>
#include <cstddef>

#define N_  16
#define P_  8732
#define C_  21
#define K_  200
#define V0_ 0.1f
#define V1_ 0.2f
#define CONF_TH 0.01f
#define NMS_TH  0.45f
#define NEG_INF (-3.0e38f)

typedef __attribute__((ext_vector_type(2))) float v2f;
typedef __attribute__((ext_vector_type(8))) float v8f;

#if defined(__has_builtin)
#if __has_builtin(__builtin_amdgcn_wmma_f32_16x16x4_f32)
#define HAVE_WMMA_F32X4 1
#endif
#endif

#define USE_ASYNC_LDS 1

// ---------------------------------------------------------------------------
// Kernel A: fuse 4-component GMM for loc (mu/var/pi), decode boxes vs priors.
// One thread per (n,p); all loads are float4 (16B) fully coalesced.
// Writes decoded / loc_aleatoric / loc_epistemic (each N*P*4 floats) to ws.
// ---------------------------------------------------------------------------
__global__ __launch_bounds__(256) void fuse_loc_kernel(
    const float* __restrict__ prior,
    const float* __restrict__ m1, const float* __restrict__ v1, const float* __restrict__ q1,
    const float* __restrict__ m2, const float* __restrict__ v2, const float* __restrict__ q2,
    const float* __restrict__ m3, const float* __restrict__ v3, const float* __restrict__ q3,
    const float* __restrict__ m4, const float* __restrict__ v4, const float* __restrict__ q4,
    float* __restrict__ decoded, float* __restrict__ locAl, float* __restrict__ locEp)
{
  const int idx = blockIdx.x * blockDim.x + threadIdx.x;
  if (idx >= N_ * P_) return;
  const int p = idx % P_;

  const float4* MU[4] = {(const float4*)m1, (const float4*)m2, (const float4*)m3, (const float4*)m4};
  const float4* VA[4] = {(const float4*)v1, (const float4*)v2, (const float4*)v3, (const float4*)v4};
  const float4* PI[4] = {(const float4*)q1, (const float4*)q2, (const float4*)q3, (const float4*)q4};

  __builtin_prefetch(&MU[0][idx + 2048], 0, 1);
  __builtin_prefetch(&PI[0][idx + 2048], 0, 1);

  float mu[4][4], pi[4][4];
  float nl[4] = {0.f, 0.f, 0.f, 0.f};
  float al[4] = {0.f, 0.f, 0.f, 0.f};
  float ep[4] = {0.f, 0.f, 0.f, 0.f};

#pragma unroll
  for (int g = 0; g < 4; ++g) {
    float4 m = MU[g][idx];
    float4 v = VA[g][idx];
    float4 q = PI[g][idx];
    mu[g][0] = m.x; mu[g][1] = m.y; mu[g][2] = m.z; mu[g][3] = m.w;
    pi[g][0] = q.x; pi[g][1] = q.y; pi[g][2] = q.z; pi[g][3] = q.w;
    nl[0] += q.x * m.x; nl[1] += q.y * m.y; nl[2] += q.z * m.z; nl[3] += q.w * m.w;
    al[0] += q.x * v.x; al[1] += q.y * v.y; al[2] += q.z * v.z; al[3] += q.w * v.w;
  }
#pragma unroll
  for (int g = 0; g < 4; ++g) {
#pragma unroll
    for (int k = 0; k < 4; ++k) {
      float d = mu[g][k] - nl[k];
      ep[k] += pi[g][k] * d * d;
    }
  }

  const float4 pr = ((const float4*)prior)[p];
  float cx = pr.x + nl[0] * V0_ * pr.z;
  float cy = pr.y + nl[1] * V0_ * pr.w;
  float w  = pr.z * __expf(nl[2] * V1_);
  float h  = pr.w * __expf(nl[3] * V1_);
  float x1 = cx - 0.5f * w;
  float y1 = cy - 0.5f * h;

  float4 box; box.x = x1; box.y = y1; box.z = x1 + w; box.w = y1 + h;
  float4 a4;  a4.x = al[0]; a4.y = al[1]; a4.z = al[2]; a4.w = al[3];
  float4 e4;  e4.x = ep[0]; e4.y = ep[1]; e4.z = ep[2]; e4.w = ep[3];
  ((float4*)decoded)[idx] = box;
  ((float4*)locAl)[idx]   = a4;
  ((float4*)locEp)[idx]   = e4;
}

// ---------------------------------------------------------------------------
// Kernel B: fuse conf GMM and transpose (N,P,C) -> (N,C,P) through LDS so the
// per-class NMS kernel reads contiguous score planes. One block per
// (n, 256-p chunk); both global read and write phases are coalesced.
// ---------------------------------------------------------------------------
#define PB_ 256
#define TS_ 257  // LDS row stride (odd) to dodge bank conflicts

__global__ __launch_bounds__(256) void fuse_conf_kernel(
    const float* __restrict__ m1, const float* __restrict__ v1, const float* __restrict__ q1,
    const float* __restrict__ m2, const float* __restrict__ v2, const float* __restrict__ q2,
    const float* __restrict__ m3, const float* __restrict__ v3, const float* __restrict__ q3,
    const float* __restrict__ m4, const float* __restrict__ v4, const float* __restrict__ q4,
    float* __restrict__ confT, float* __restrict__ calT, float* __restrict__ cepT)
{
  __shared__ float tC[C_ * TS_];
  __shared__ float tA[C_ * TS_];
  __shared__ float tE[C_ * TS_];

  const int nBlkP = (P_ + PB_ - 1) / PB_;   // 35
  const int nb = blockIdx.x / nBlkP;
  const int pb = (blockIdx.x % nBlkP) * PB_;
  const int pcnt = min(PB_, P_ - pb);
  const int nelem = pcnt * C_;
  const size_t base = (size_t)nb * P_ * C_ + (size_t)pb * C_;

  const float* MU[4] = {m1, m2, m3, m4};
  const float* VA[4] = {v1, v2, v3, v4};
  const float* PI[4] = {q1, q2, q3, q4};

  __builtin_prefetch(MU[0] + base + nelem, 0, 1);
  __builtin_prefetch(VA[0] + base + nelem, 0, 1);
  __builtin_prefetch(PI[0] + base + nelem, 0, 1);

  for (int e = threadIdx.x; e < nelem; e += 256) {
    float mu[4], va[4], pi[4];
#pragma unroll
    for (int g = 0; g < 4; ++g) {
      mu[g] = MU[g][base + e];
      va[g] = VA[g][base + e];
      pi[g] = PI[g][base + e];
    }
    float nc = 0.f, ca = 0.f, ce = 0.f;
#pragma unroll
    for (int g = 0; g < 4; ++g) { nc += pi[g] * mu[g]; ca += pi[g] * va[g]; }
#pragma unroll
    for (int g = 0; g < 4; ++g) { float d = mu[g] - nc; ce += pi[g] * d * d; }
    int pl = e / C_;
    int c  = e - pl * C_;
    tC[c * TS_ + pl] = nc;
    tA[c * TS_ + pl] = ca;
    tE[c * TS_ + pl] = ce;
  }
  __syncthreads();

  if (threadIdx.x < pcnt) {
    for (int c = 0; c < C_; ++c) {
      size_t o = ((size_t)nb * C_ + c) * (size_t)P_ + pb + threadIdx.x;
      confT[o] = tC[c * TS_ + threadIdx.x];
      calT[o]  = tA[c * TS_ + threadIdx.x];
      cepT[o]  = tE[c * TS_ + threadIdx.x];
    }
  }
}

// ---------------------------------------------------------------------------
// Zero-fill of d_out (harness poisons it; most rows must be exactly 0).
// ---------------------------------------------------------------------------
__global__ void zero_kernel(float* __restrict__ out, int n)
{
  int i = blockIdx.x * blockDim.x + threadIdx.x;
  if (i < n) out[i] = 0.f;
}

// ---------------------------------------------------------------------------
// Kernel C: one 256-thread block per (image, fg class).
//  phase 1: async-DMA the 8732-float score plane into LDS, threshold-mask
//  phase 2: iterative top-200 argmax (wave32 shuffle reduce + 8-wave combine)
//  phase 3: gather boxes/areas for the 200 candidates
//  phase 4: 200x200 IoU matrix in LDS; area[i]+area[j] term per 16x16 tile
//           via V_WMMA_F32_16X16X4_F32 (A=[area,1,0,0], B=[1,area,0,0]^T)
//  phase 5: greedy NMS (sequential over j, parallel over i<j)
//  phase 6: stable compaction + 15-float row writeback
// ---------------------------------------------------------------------------
#define PADK 208          // K_ padded to 16 for WMMA tiling
#define NTILE 169         // 13*13 tiles of 16x16
#define IOUW 209

__global__ __launch_bounds__(256) void nms_kernel(
    const float* __restrict__ confT, const float* __restrict__ calT,
    const float* __restrict__ cepT,  const float* __restrict__ decoded,
    const float* __restrict__ locAl, const float* __restrict__ locEp,
    float* __restrict__ out)
{
  __shared__ __align__(16) float scoreBuf[8736];
  __shared__ float iouLds[PADK * IOUW];
  __shared__ float bx1[PADK], by1[PADK], bx2[PADK], by2[PADK], areaLds[PADK];
  __shared__ float selVal[PADK];
  __shared__ int   selIdx[PADK];
  __shared__ int   keepLds[PADK];
  __shared__ int   posLds[PADK];
  __shared__ float redV[8];
  __shared__ int   redI[8];
  __shared__ int   flag;

  const int tid  = threadIdx.x;
  const int lane = tid & 31;
  const int wid  = tid >> 5;
  const int n    = blockIdx.x / 20;
  const int ci   = blockIdx.x % 20;
  const int cls  = ci + 1;                       // fg class -> output channel
  const float* plane = confT + ((size_t)n * C_ + cls) * (size_t)P_;

  // ---- phase 1: scores -> LDS ----
#if USE_ASYNC_LDS
  for (int k = tid; k < 2184; k += 256) {        // 2184 * 16B = 8736 floats
    uint32_t lds = (uint32_t)(uintptr_t)(&scoreBuf[k * 4]);
    const float* g = plane + (size_t)k * 4;
    asm volatile("global_load_async_to_lds_b128 %0, %1, off"
                 :: "v"(lds), "v"(g) : "memory");
  }
  asm volatile("s_wait_asynccnt 0" ::: "memory");
  __syncthreads();
#pragma unroll
  for (int k = 0; k < 35; ++k) {
    int i = tid + (k << 8);
    if (i < 8736) {
      float s = scoreBuf[i];
      scoreBuf[i] = (i < P_ && s > CONF_TH) ? s : NEG_INF;
    }
  }
#else
#pragma unroll
  for (int k = 0; k < 35; ++k) {
    int i = tid + (k << 8);
    if (i < 8736) {
      float s = (i < P_) ? plane[i] : NEG_INF;
      scoreBuf[i] = (s > CONF_TH) ? s : NEG_INF;
    }
  }
#endif
  __syncthreads();

  // ---- phase 2: iterative top-K argmax ----
  for (int it = 0; it < K_; ++it) {
    float v = NEG_INF;
    int   ii = 0;
#pragma unroll
    for (int k = 0; k < 35; ++k) {
      int i = tid + (k << 8);
      if (i < 8736) {
        float s = scoreBuf[i];
        if (s > v) { v = s; ii = i; }
      }
    }
    for (int off = 16; off > 0; off >>= 1) {
      float ov = __shfl_down(v, (unsigned)off, 32);
      int   oi = __shfl_down(ii, (unsigned)off, 32);
      if (ov > v || (ov == v && oi < ii)) { v = ov; ii = oi; }
    }
    if (lane == 0) { redV[wid] = v; redI[wid] = ii; }
    __syncthreads();
    if (tid == 0) {
      float bv = redV[0]; int bi = redI[0];
#pragma unroll
      for (int w = 1; w < 8; ++w) {
        float wv = redV[w]; int wi = redI[w];
        if (wv > bv || (wv == bv && wi < bi)) { bv = wv; bi = wi; }
      }
      selVal[it] = bv; selIdx[it] = bi;
      scoreBuf[bi] = NEG_INF;                    // remove winner
    }
    __syncthreads();
  }

  // ---- phase 3: gather candidate boxes ----
  if (tid < PADK) {
    if (tid < K_) {
      int idx = selIdx[tid];
      float4 b = ((const float4*)decoded)[(size_t)n * P_ + idx];
      bx1[tid] = b.x; by1[tid] = b.y; bx2[tid] = b.z; by2[tid] = b.w;
      areaLds[tid] = (b.z - b.x) * (b.w - b.y);
    } else {
      bx1[tid] = by1[tid] = bx2[tid] = by2[tid] = 0.f;
      areaLds[tid] = 0.f;
      selVal[tid] = NEG_INF;
      selIdx[tid] = 0;
    }
    keepLds[tid] = 0;
  }
  __syncthreads();

  // ---- phase 4: IoU matrix; area-sum per tile via WMMA f32 16x16x4 ----
  for (int t = wid; t < NTILE; t += 8) {
    int ti = t / 13, tj = t - ti * 13;
    int i0 = ti << 4, j0 = tj << 4;
    int ls = lane & 15;
    float av = areaLds[i0 + ls];
    float bv = areaLds[j0 + ls];
    bool lo = (lane < 16);

    v8f D;
#if defined(HAVE_WMMA_F32X4)
    // A (16x4 f32): lanes 0-15 hold M=lane {K0,K1}; lanes 16-31 hold {K2,K3}=0
    v2f A; A.x = lo ? av  : 0.f; A.y = lo ? 1.f : 0.f;
    // B (4x16 f32): vgpr0 = K0 (lanes<16) / K2; vgpr1 = K1 / K3
    v2f B; B.x = lo ? 1.f : 0.f; B.y = lo ? bv  : 0.f;
    v8f Cc = {0.f, 0.f, 0.f, 0.f, 0.f, 0.f, 0.f, 0.f};
    D = __builtin_amdgcn_wmma_f32_16x16x4_f32(false, A, false, B,
                                              (short)0, Cc, false, false);
#else
    int ibf = i0 + ((lane >> 4) << 3);
#pragma unroll
    for (int r = 0; r < 8; ++r) D[r] = areaLds[ibf + r] + areaLds[j0 + (lane & 15)];
#endif
    // D layout: vgpr r -> M = r (+8 for lanes 16-31), N = lane&15
    int ib = i0 + ((lane >> 4) << 3);
    int j  = j0 + (lane & 15);
    float xj1 = bx1[j], yj1 = by1[j], xj2 = bx2[j], yj2 = by2[j];
#pragma unroll
    for (int r = 0; r < 8; ++r) {
      int i = ib + r;
      float xx1 = fmaxf(bx1[i], xj1);
      float yy1 = fmaxf(by1[i], yj1);
      float xx2 = fminf(bx2[i], xj2);
      float yy2 = fminf(by2[i], yj2);
      float inter = fmaxf(xx2 - xx1, 0.f) * fmaxf(yy2 - yy1, 0.f);
      iouLds[i * IOUW + j] = inter / (D[r] - inter);
    }
  }
  __syncthreads();

  // ---- phase 5: greedy NMS ----
  for (int j = 0; j < K_; ++j) {
    if (tid == 0) flag = 0;
    __syncthreads();
    if (tid < j && keepLds[tid] && iouLds[tid * IOUW + j] > NMS_TH) flag = 1;
    __syncthreads();
    if (tid == 0) keepLds[j] = (selVal[j] > CONF_TH && flag == 0) ? 1 : 0;
    __syncthreads();
  }

  // ---- phase 6: stable compaction + writeback ----
  if (tid == 0) {
    int c = 0;
    for (int j = 0; j < K_; ++j) { posLds[j] = c; c += keepLds[j]; }
  }
  __syncthreads();

  if (tid < K_ && keepLds[tid]) {
    int idx = selIdx[tid];
    float* row = out + ((size_t)(n * C_ + cls) * K_ + posLds[tid]) * 15;
    row[0] = selVal[tid];
    row[1] = bx1[tid]; row[2] = by1[tid]; row[3] = bx2[tid]; row[4] = by2[tid];
    float4 la = ((const float4*)locAl)[(size_t)n * P_ + idx];
    float4 le = ((const float4*)locEp)[(size_t)n * P_ + idx];
    row[5] = la.x; row[6]  = la.y; row[7]  = la.z; row[8]  = la.w;
    row[9] = le.x; row[10] = le.y; row[11] = le.z; row[12] = le.w;
    size_t po = ((size_t)n * C_ + cls) * (size_t)P_ + idx;
    row[13] = calT[po];
    row[14] = cepT[po];
  }
}

// ---------------------------------------------------------------------------
extern "C" void kernel_launch(void* const* d_in, const int* in_sizes, int n_in,
                              void* d_out, int out_size, void* d_ws, size_t ws_size,
                              hipStream_t stream)
{
  (void)in_sizes; (void)n_in; (void)ws_size;

  const float* prior = (const float*)d_in[0];
  const float* L[12];
  const float* Cc[12];
  for (int i = 0; i < 12; ++i) L[i]  = (const float*)d_in[1 + i];
  for (int i = 0; i < 12; ++i) Cc[i] = (const float*)d_in[13 + i];

  float* ws = (float*)d_ws;
  const size_t NP4 = (size_t)N_ * P_ * 4;      // 558,848
  const size_t NCP = (size_t)N_ * C_ * P_;     // 2,933,952
  float* decoded = ws;
  float* locAl   = ws + NP4;
  float* locEp   = ws + 2 * NP4;
  float* confT   = ws + 3 * NP4;
  float* calT    = confT + NCP;
  float* cepT    = calT + NCP;

  float* out = (float*)d_out;

  // A: loc mixture + decode
  int gridA = (N_ * P_ + 255) / 256;
  fuse_loc_kernel<<<gridA, 256, 0, stream>>>(
      prior,
      L[0], L[1], L[2], L[3], L[4], L[5], L[6], L[7], L[8], L[9], L[10], L[11],
      decoded, locAl, locEp);

  // B: conf mixture + transpose to (N,C,P)
  int gridB = N_ * ((P_ + PB_ - 1) / PB_);
  fuse_conf_kernel<<<gridB, 256, 0, stream>>>(
      Cc[0], Cc[1], Cc[2], Cc[3], Cc[4], Cc[5], Cc[6], Cc[7], Cc[8], Cc[9], Cc[10], Cc[11],
      confT, calT, cepT);

  // zero the output (bg class + all non-kept rows must be exactly 0)
  zero_kernel<<<(out_size + 255) / 256, 256, 0, stream>>>(out, out_size);

  // C: top-k + IoU(WMMA) + greedy NMS + writeback, one block per (n, class)
  nms_kernel<<<N_ * 20, 256, 0, stream>>>(confT, calT, cepT,
                                          decoded, locAl, locEp, out);
}